// MergedEmbeddingBag_16527034155603
// MI455X (gfx1250) — compile-verified
//
#include <hip/hip_runtime.h>
#include <stdint.h>

// ---- problem constants (from reference) ----
#define NT 26      // tables
#define NR 100000  // rows per table
#define ND 128     // embedding dim
#define NB 4096    // batch (bags per table)
#define NL 20      // bag size

#define WAVES_PER_BLOCK 8
#define CHUNK 8    // rows per TDM gather op (32-bit index mode supports up to 8)

typedef unsigned int u32x4 __attribute__((ext_vector_type(4)));
typedef int          i32x8 __attribute__((ext_vector_type(8)));
typedef int          i32x4 __attribute__((ext_vector_type(4)));
typedef float        v4f   __attribute__((ext_vector_type(4)));

// Issue one TDM gather: pull `nidx` rows (row indices idx[0..nidx-1]) of
// ND fp32 elements from the table at `table_base` into LDS at byte address
// `lds_byte_addr` (rows packed consecutively, 512 B apart).
// Descriptor bit layout per CDNA5 ISA.
__device__ __forceinline__ void tdm_gather_rows(uint32_t lds_byte_addr,
                                                uint64_t table_base,
                                                const int* idx, int nidx) {
  // ---- D# group 0 (128 bits) ----
  u32x4 g0;
  g0.x = 1u              // count = 1 (one valid descriptor)
       | (1u << 30)      // gather_index_size = 1 -> 32-bit indices
       | (1u << 31);     // gather_mode = 1
  g0.y = lds_byte_addr;                                  // lds_addr
  g0.z = (uint32_t)(table_base & 0xffffffffu);           // global_addr[31:0]
  g0.w = (uint32_t)((table_base >> 32) & 0x01ffffffu)    // global_addr[56:32]
       | (2u << 30);                                     // type = 2 ("image")

  // ---- D# group 1 (256 bits) ----
  i32x8 g1;
  g1[0] = (2 << 16);                         // workgroup_mask=0, data_size=2 (4B)
  g1[1] = (ND << 16);                        // tensor_dim0[15:0] = 128
  g1[2] = (int)((NR & 0xffff) << 16);        // tensor_dim0 hi=0 | tensor_dim1 lo
  g1[3] = (NR >> 16) | (ND << 16);           // tensor_dim1 hi | tile_dim0 = 128
  g1[4] = nidx;                              // tile_dim1 = #valid gather indices
  g1[5] = ND;                                // tensor_dim0_stride = 128 elems
  g1[6] = 0;                                 // (tensor_dim1_stride ignored in gather)
  g1[7] = 0;

  // ---- D# groups 2/3: 32-bit row indices (up to 8) ----
  i32x4 g2, g3;
  g2[0] = idx[0];
  g2[1] = (nidx > 1) ? idx[1] : 0;
  g2[2] = (nidx > 2) ? idx[2] : 0;
  g2[3] = (nidx > 3) ? idx[3] : 0;
  g3[0] = (nidx > 4) ? idx[4] : 0;
  g3[1] = (nidx > 5) ? idx[5] : 0;
  g3[2] = (nidx > 6) ? idx[6] : 0;
  g3[3] = (nidx > 7) ? idx[7] : 0;

#if __has_include(<hip/amd_detail/amd_gfx1250_TDM.h>)
  // amdgpu-toolchain (clang-23): 6-arg form
  i32x8 zero8 = 0;
  __builtin_amdgcn_tensor_load_to_lds(g0, g1, g2, g3, zero8, 0);
#else
  // ROCm 7.2 (clang-22): 5-arg form
  __builtin_amdgcn_tensor_load_to_lds(g0, g1, g2, g3, 0);
#endif
}

__global__ __launch_bounds__(256) void
MergedEmbeddingBag_16527034155603_kernel(const float* __restrict__ W,
                                         const int* __restrict__ indices,
                                         float* __restrict__ out) {
  // Per-wave double-buffered staging: 2 x 8 rows x 512 B = 8 KB per wave,
  // 64 KB per 8-wave block.
  __shared__ float stage[WAVES_PER_BLOCK][2][CHUNK][ND];

  const int lane = threadIdx.x & 31;
  const int wave = threadIdx.x >> 5;
  const int t    = blockIdx.y;                                // table
  const int bag  = blockIdx.x * WAVES_PER_BLOCK + wave;       // bag in batch

  // 20 bag indices; uniform per wave -> force into SGPRs for the descriptor.
  const int* bagIdx = indices + ((size_t)t * NB + (size_t)bag) * NL;
  int ind[NL];
#pragma unroll
  for (int i = 0; i < NL; ++i)
    ind[i] = __builtin_amdgcn_readfirstlane(bagIdx[i]);

  const uint64_t tableBase =
      (uint64_t)(uintptr_t)(W + (size_t)t * (size_t)NR * ND);
  // Generic pointer to LDS: low 32 bits are the LDS byte address.
  const uint32_t lds0 = (uint32_t)(uintptr_t)&stage[wave][0][0][0];
  const uint32_t lds1 = (uint32_t)(uintptr_t)&stage[wave][1][0][0];

  // Software pipeline: issue chunk0 + chunk1, consume chunk0, issue chunk2,
  // consume chunk1, consume chunk2.  TDM ops from one wave complete in order.
  tdm_gather_rows(lds0, tableBase, &ind[0], 8);
  tdm_gather_rows(lds1, tableBase, &ind[8], 8);

  v4f acc = (v4f)0.0f;
  const v4f* b0 = (const v4f*)&stage[wave][0][0][0];
  const v4f* b1 = (const v4f*)&stage[wave][1][0][0];

  __builtin_amdgcn_s_wait_tensorcnt((short)1);   // chunk0 landed in LDS
#pragma unroll
  for (int r = 0; r < 8; ++r)
    acc += b0[r * 32 + lane];                    // row = 32 lanes x 16 B

  // Buffer-0 reuse: TDM writes are unordered vs DS reads, so drain DS first.
  asm volatile("s_wait_dscnt 0" ::: "memory");
  tdm_gather_rows(lds0, tableBase, &ind[16], 4);

  __builtin_amdgcn_s_wait_tensorcnt((short)1);   // chunk1 landed
#pragma unroll
  for (int r = 0; r < 8; ++r)
    acc += b1[r * 32 + lane];

  __builtin_amdgcn_s_wait_tensorcnt((short)0);   // chunk2 landed
#pragma unroll
  for (int r = 0; r < 4; ++r)
    acc += b0[r * 32 + lane];

  // MODES = [0,1,0,1,...]: odd tables are MEAN (x 1/20), even are SUM.
  const float scale = (t & 1) ? (1.0f / (float)NL) : 1.0f;
  acc *= scale;

  v4f* outp = (v4f*)(out + ((size_t)t * NB + (size_t)bag) * ND);
  outp[lane] = acc;
}

extern "C" void kernel_launch(void* const* d_in, const int* in_sizes, int n_in,
                              void* d_out, int out_size, void* d_ws, size_t ws_size,
                              hipStream_t stream) {
  const float* W   = (const float*)d_in[0];
  const int* idx   = (const int*)d_in[1];
  float* out       = (float*)d_out;

  // blockIdx.y = table (slow dimension) so all blocks of one table run
  // near-simultaneously and share its rows in the 192 MB L2.
  dim3 grid(NB / WAVES_PER_BLOCK, NT);
  MergedEmbeddingBag_16527034155603_kernel<<<grid, 256, 0, stream>>>(W, idx, out);
}